// QConv2D_27187142984003
// MI455X (gfx1250) — compile-verified
//
#include <hip/hip_runtime.h>
#include <math.h>

typedef __attribute__((ext_vector_type(2))) float v2f;
typedef __attribute__((ext_vector_type(8))) float v8f;

#define NQ      4
#define NST     16
#define HO      127
#define WO      127
#define NPB     (HO * WO)        // 16129 patches per batch image
#define BATCH   32
#define NTOT    (BATCH * NPB)    // 516128 total patches
#define IW      128
#define NBASIS  81               // 3^4 multilinear basis {1, cos, sin}^4
#define KPAD    84               // pad K to multiple of 4 for 16x16x4 WMMA
#define NCHUNK  (KPAD / 4)       // 21 WMMA K-chunks

// ---------------------------------------------------------------------------
// Stage 1: recover coefficient tensor C[81][4] from params by exact
// 3-point-per-axis interpolation of the 4-qubit circuit output.
// ---------------------------------------------------------------------------

__device__ __forceinline__ void gate_rx(float* re, float* im, int w, float th) {
  float c = cosf(0.5f * th), s = sinf(0.5f * th);
  int mk = 8 >> w;  // wire 0 is MSB of the 4-bit basis index
  #pragma unroll
  for (int idx = 0; idx < NST; ++idx) {
    if (!(idx & mk)) {
      int jdx = idx | mk;
      float r0 = re[idx], i0 = im[idx], r1 = re[jdx], i1 = im[jdx];
      // new0 = c*s0 - i*s*s1 ; new1 = c*s1 - i*s*s0
      re[idx] = c * r0 + s * i1;
      im[idx] = c * i0 - s * r1;
      re[jdx] = c * r1 + s * i0;
      im[jdx] = c * i1 - s * r0;
    }
  }
}

__device__ __forceinline__ void gate_rz(float* re, float* im, int w, float th) {
  float c = cosf(0.5f * th), s = sinf(0.5f * th);
  int mk = 8 >> w;
  #pragma unroll
  for (int idx = 0; idx < NST; ++idx) {
    float r = re[idx], i = im[idx];
    if (idx & mk) { re[idx] = c * r - s * i; im[idx] = c * i + s * r; }  // e^{+i th/2}
    else          { re[idx] = c * r + s * i; im[idx] = c * i - s * r; }  // e^{-i th/2}
  }
}

__device__ __forceinline__ void gate_cnot(float* re, float* im, int cw, int tw) {
  int mc = 8 >> cw, mt = 8 >> tw;
  #pragma unroll
  for (int idx = 0; idx < NST; ++idx) {
    if ((idx & mc) && !(idx & mt)) {
      int jdx = idx | mt;
      float tr = re[idx]; re[idx] = re[jdx]; re[jdx] = tr;
      float ti = im[idx]; im[idx] = im[jdx]; im[jdx] = ti;
    }
  }
}

__global__ __launch_bounds__(128)
void qconv_coeff_kernel(const float* __restrict__ params, float* __restrict__ Cws) {
  __shared__ float Y[NBASIS][NQ];
  const int tid = threadIdx.x;

  if (tid < NBASIS) {
    const int t0 = tid % 3, t1 = (tid / 3) % 3, t2 = (tid / 9) % 3, t3 = tid / 27;
    const float HPI = 1.57079632679489662f;
    float ang[NQ] = { t0 * HPI, t1 * HPI, t2 * HPI, t3 * HPI };

    float re[NST], im[NST];
    #pragma unroll
    for (int i = 0; i < NST; ++i) { re[i] = 0.f; im[i] = 0.f; }
    re[0] = 1.f;

    // AngleEmbedding: RX(x_w) on wire w
    #pragma unroll
    for (int w = 0; w < NQ; ++w) gate_rx(re, im, w, ang[w]);

    // 2 layers: RX, RZ per wire, then CNOT(control=w+1, target=w)
    #pragma unroll
    for (int l = 0; l < 2; ++l) {
      #pragma unroll
      for (int w = 0; w < NQ; ++w) {
        gate_rx(re, im, w, params[l * 8 + w * 2 + 0]);
        gate_rz(re, im, w, params[l * 8 + w * 2 + 1]);
      }
      #pragma unroll
      for (int w = 0; w < NQ - 1; ++w) gate_cnot(re, im, w + 1, w);
    }

    // <Z_w> = P(bit_w = 0) - P(bit_w = 1)
    #pragma unroll
    for (int w = 0; w < NQ; ++w) {
      int mk = 8 >> w;
      float z = 0.f;
      #pragma unroll
      for (int i = 0; i < NST; ++i) {
        float pr = re[i] * re[i] + im[i] * im[i];
        z += (i & mk) ? -pr : pr;
      }
      Y[tid][w] = z;
    }
  }

  // Interpolate samples -> coefficients, axis by axis.
  // y(x) = c0 + c1*cos x + c2*sin x sampled at x = 0, pi/2, pi:
  //   c0 = (y0+y2)/2 ; c1 = (y0-y2)/2 ; c2 = y1 - c0
  const int pow3[4] = {1, 3, 9, 27};
  for (int v = 0; v < 4; ++v) {
    __syncthreads();
    if (tid < 27 * NQ) {
      int w = tid & 3;
      int g = tid >> 2;            // 0..26: combinations of the other 3 digits
      int s = pow3[v];
      int low = g % s;
      int high = g / s;
      int k = low + high * 3 * s;  // digit v == 0 position
      float y0 = Y[k][w], y1 = Y[k + s][w], y2 = Y[k + 2 * s][w];
      float c0 = 0.5f * (y0 + y2);
      float c1 = 0.5f * (y0 - y2);
      float c2 = y1 - c0;
      Y[k][w] = c0; Y[k + s][w] = c1; Y[k + 2 * s][w] = c2;
    }
  }
  __syncthreads();

  // Write padded coefficient matrix Cws[KPAD][NQ] (rows 81..83 zero).
  for (int idx = tid; idx < KPAD * NQ; idx += blockDim.x) {
    int k = idx >> 2, w = idx & 3;
    Cws[idx] = (k < NBASIS) ? Y[k][w] : 0.f;
  }
}

// ---------------------------------------------------------------------------
// Stage 2: per-patch feature map phi (81 products of {1,cos,sin}) + GEMM
// (16x84) x (84x16, only 4 columns live) via V_WMMA_F32_16X16X4_F32.
// 128 threads = 4 waves; each wave owns 32 patches -> two 16x16 D tiles.
// ---------------------------------------------------------------------------

__global__ __launch_bounds__(128)
void qconv_main_kernel(const float* __restrict__ x,
                       const float* __restrict__ Cws,
                       float* __restrict__ out) {
  __shared__ float PHI[4][32][85];  // stride 85 -> gcd(85,64)=1, conflict-free

  const int lane = threadIdx.x & 31;
  const int wave = threadIdx.x >> 5;
  const int wavePatchBase = blockIdx.x * 128 + wave * 32;

  // ---- per-lane patch -> feature vector phi ----
  int p = wavePatchBase + lane;
  if (p >= NTOT) p = NTOT - 1;  // clamp; stores are guarded later
  int b   = p / NPB;
  int rem = p - b * NPB;
  int i   = rem / WO;
  int j   = rem - i * WO;
  const float* src = x + (b * IW + i) * IW + j;
  float x0 = src[0];
  float x1 = src[1];
  float x2 = src[IW];
  float x3 = src[IW + 1];

  float u0[3] = {1.f, cosf(x0), sinf(x0)};
  float u1[3] = {1.f, cosf(x1), sinf(x1)};
  float u2[3] = {1.f, cosf(x2), sinf(x2)};
  float u3[3] = {1.f, cosf(x3), sinf(x3)};

  float w01[9], w23[9];
  #pragma unroll
  for (int bq = 0; bq < 3; ++bq)
    #pragma unroll
    for (int aq = 0; aq < 3; ++aq) {
      w01[aq + 3 * bq] = u0[aq] * u1[bq];  // index = t0 + 3*t1
      w23[aq + 3 * bq] = u2[aq] * u3[bq];  // index = t2 + 3*t3
    }

  float* phi = &PHI[wave][lane][0];
  #pragma unroll
  for (int hq = 0; hq < 9; ++hq)
    #pragma unroll
    for (int lq = 0; lq < 9; ++lq)
      phi[lq + 9 * hq] = w01[lq] * w23[hq];  // k = t0+3t1+9t2+27t3
  phi[81] = 0.f; phi[82] = 0.f; phi[83] = 0.f;

  // ---- load B tiles (coefficients) into registers ----
  // B 4x16 f32 layout: lanes 0-15 -> N=lane, VGPR0=K0, VGPR1=K1;
  //                    lanes 16-31 -> N=lane-16, VGPR0=K2, VGPR1=K3.
  const int n  = lane & 15;
  const int kb = (lane >> 4) << 1;  // 0 or 2
  v2f Br[NCHUNK];
  #pragma unroll
  for (int c = 0; c < NCHUNK; ++c) {
    float bx = 0.f, by = 0.f;
    if (n < NQ) {
      bx = Cws[(4 * c + kb    ) * NQ + n];
      by = Cws[(4 * c + kb + 1) * NQ + n];
    }
    Br[c].x = bx;
    Br[c].y = by;
  }

  __syncthreads();

  // ---- WMMA accumulation over 21 K-chunks, two M tiles per wave ----
  // A 16x4 f32 layout: lanes 0-15 -> M=lane, VGPR0=K0, VGPR1=K1;
  //                    lanes 16-31 -> M=lane-16, VGPR0=K2, VGPR1=K3.
  v8f acc0 = {};
  v8f acc1 = {};
  const int m = lane & 15;
  #pragma unroll
  for (int c = 0; c < NCHUNK; ++c) {
    v2f a0, a1;
    a0.x = PHI[wave][m     ][4 * c + kb];
    a0.y = PHI[wave][m     ][4 * c + kb + 1];
    a1.x = PHI[wave][m + 16][4 * c + kb];
    a1.y = PHI[wave][m + 16][4 * c + kb + 1];
    acc0 = __builtin_amdgcn_wmma_f32_16x16x4_f32(false, a0, false, Br[c],
                                                 (short)0, acc0, false, false);
    acc1 = __builtin_amdgcn_wmma_f32_16x16x4_f32(false, a1, false, Br[c],
                                                 (short)0, acc1, false, false);
  }

  // ---- scatter D columns 0..3 to output (B, 4, 127, 127) ----
  // D layout: VGPR r, lane l -> D[M = r + 8*(l>>4)][N = l&15]
  if (n < NQ) {
    const int half = lane >> 4;
    #pragma unroll
    for (int r = 0; r < 8; ++r) {
      int pp = wavePatchBase + r + 8 * half;           // tile 0
      if (pp < NTOT) {
        int bb = pp / NPB;
        int rr = pp - bb * NPB;
        out[(bb * NQ + n) * NPB + rr] = acc0[r];
      }
      pp += 16;                                        // tile 1
      if (pp < NTOT) {
        int bb = pp / NPB;
        int rr = pp - bb * NPB;
        out[(bb * NQ + n) * NPB + rr] = acc1[r];
      }
    }
  }
}

// ---------------------------------------------------------------------------

extern "C" void kernel_launch(void* const* d_in, const int* in_sizes, int n_in,
                              void* d_out, int out_size, void* d_ws, size_t ws_size,
                              hipStream_t stream) {
  const float* x      = (const float*)d_in[0];  // (32,1,128,128) f32
  const float* params = (const float*)d_in[1];  // (2,4,2) f32
  float* out = (float*)d_out;                   // (32,4,127,127) f32
  float* Cws = (float*)d_ws;                    // 84*4 f32 coefficients

  qconv_coeff_kernel<<<1, 128, 0, stream>>>(params, Cws);

  const int nblocks = (NTOT + 127) / 128;       // 4033
  qconv_main_kernel<<<nblocks, 128, 0, stream>>>(x, Cws, out);
}